// GATLayer_32719060861314
// MI455X (gfx1250) — compile-verified
//
#include <hip/hip_runtime.h>
#include <math.h>

typedef __attribute__((ext_vector_type(2))) float v2f;
typedef __attribute__((ext_vector_type(8))) float v8f;

#define DCH   128
#define NHEAD 4

// ---------------- fast fp32 global atomic add ----------------------------------
__device__ __forceinline__ void atomAddF32(float* p, float v) {
#if defined(__HIP_DEVICE_COMPILE__)
    unsafeAtomicAdd(p, v);    // lowers to global_atomic_add_f32 on CDNA
#else
    atomicAdd(p, v);
#endif
}

// ---------------- helpers: order-preserving float<->int for atomicMax ----------
__device__ __forceinline__ int enc_f32(float f) {
    int b = __float_as_int(f);
    return (b >= 0) ? b : (b ^ 0x7FFFFFFF);
}
__device__ __forceinline__ float dec_f32(int k) {
    return __int_as_float((k >= 0) ? k : (k ^ 0x7FFFFFFF));
}

// ---------------- init: zero output & denom, seed segment-max with -inf --------
__global__ __launch_bounds__(256)
void init_kernel(int* __restrict__ mbits, float* __restrict__ denom,
                 float* __restrict__ out, int n)
{
    const int i = blockIdx.x * 256 + threadIdx.x;
    if (i < n * NHEAD) {
        mbits[i] = (int)0x807FFFFF;   // enc_f32(-inf)
        denom[i] = 0.f;
    }
    if (i < n * DCH) out[i] = 0.f;
}

// ---------------- fused K/Q/V projection GEMM: out = x @ W^T + b ---------------
// grid: (ceil(n/128), 1, 3)  block: 256 (8 waves). Wave w computes rows
// [blk*128 + 16w, +16) x all 128 cols with fp32 WMMA 16x16x4.
//
// LDS holds B = W^T in a PAIR-INTERLEAVED layout so each lane's B fragment
// (rows k0+2h and k0+2h+1 at one column) is one contiguous 8-byte ds_load_b64:
//   lW2[(d>>1)*256 + j*2 + (d&1)] = W[j][d]   (exactly 64 KB)
__global__ __launch_bounds__(256)
void proj_gemm(const float* __restrict__ x,
               const float* __restrict__ Wk, const float* __restrict__ bk,
               const float* __restrict__ Wq, const float* __restrict__ bq,
               const float* __restrict__ Wv, const float* __restrict__ bv,
               float* __restrict__ ok, float* __restrict__ oq, float* __restrict__ ov,
               int n)
{
    const float* __restrict__ W;
    const float* __restrict__ bias;
    float* __restrict__ out;
    if (blockIdx.z == 0)      { W = Wk; bias = bk; out = ok; }
    else if (blockIdx.z == 1) { W = Wq; bias = bq; out = oq; }
    else                      { W = Wv; bias = bv; out = ov; }

    __shared__ float lW2[DCH * DCH];

    const int tid  = threadIdx.x;
    const int wave = tid >> 5;
    const int lane = tid & 31;
    const int m    = lane & 15;   // row (A) / col (B) within 16-tile
    const int half = lane >> 4;   // K-pair select per ISA fragment layout

    for (int i = tid; i < DCH * (DCH / 4); i += 256) {
        const int j  = i >> 5;          // W row == output column
        const int d4 = (i & 31) * 4;    // element within row
        const float4 w4 = *(const float4*)(W + j * DCH + d4);
        *(v2f*)&lW2[((d4 >> 1) + 0) * 256 + j * 2] = (v2f){w4.x, w4.y};
        *(v2f*)&lW2[((d4 >> 1) + 1) * 256 + j * 2] = (v2f){w4.z, w4.w};
    }
    __syncthreads();

    // Clamp instead of predicating: rows >= n read row n-1 (harmless garbage),
    // keeps the k-loop free of EXEC save/restore; output store is guarded.
    const int arow  = blockIdx.x * DCH + wave * 16 + m;
    const int arowc = (arow < n) ? arow : (n - 1);
    const float* xrow = x + (size_t)arowc * DCH;

    v8f acc[8];
#pragma unroll
    for (int t = 0; t < 8; ++t) acc[t] = (v8f){0.f,0.f,0.f,0.f,0.f,0.f,0.f,0.f};

    for (int k0 = 0; k0 < DCH; k0 += 4) {
        // A 16x4 fp32 fragment: lanes0-15 hold K=k0,k0+1; lanes16-31 K=k0+2,k0+3
        const v2f a = *(const v2f*)(xrow + k0 + half * 2);
        const int prow = (k0 >> 1) + half;    // pair index for rows k0+2h, k0+2h+1
#pragma unroll
        for (int t = 0; t < 8; ++t) {
            // B 4x16 fragment: VGPR0 rows {K0|K2}, VGPR1 rows {K1|K3}, N = lane&15
            const int col = t * 16 + m;
            const v2f bf = *(const v2f*)&lW2[prow * 256 + col * 2];
            acc[t] = __builtin_amdgcn_wmma_f32_16x16x4_f32(
                false, a, false, bf, (short)0, acc[t], false, false);
        }
    }

    // C/D layout: VGPR r -> (M=r, lanes0-15) and (M=r+8, lanes16-31), N=lane&15
    const int rowbase = blockIdx.x * DCH + wave * 16 + half * 8;
#pragma unroll
    for (int t = 0; t < 8; ++t) {
        const int col = t * 16 + m;
        const float bb = bias[col];
#pragma unroll
        for (int r = 0; r < 8; ++r) {
            const int grow = rowbase + r;
            if (grow < n) out[(size_t)grow * DCH + col] = acc[t][r] + bb;
        }
    }
}

// ---------------- per-edge per-head scores + segment max -----------------------
// One wave per edge; lane l loads float4 covering dims [4l,4l+3]; head = l>>3.
__global__ __launch_bounds__(256)
void edge_scores_kernel(const float* __restrict__ kf, const float* __restrict__ qf,
                        const int* __restrict__ src, const int* __restrict__ dst,
                        float* __restrict__ scores, int* __restrict__ mbits, int ne)
{
    const int e = blockIdx.x * 8 + (threadIdx.x >> 5);
    if (e >= ne) return;
    const int lane = threadIdx.x & 31;
    const int s = src[e], d = dst[e];
    const float4 kk = *(const float4*)(kf + (size_t)s * DCH + lane * 4);
    const float4 qq = *(const float4*)(qf + (size_t)d * DCH + lane * 4);
    float p = kk.x*qq.x + kk.y*qq.y + kk.z*qq.z + kk.w*qq.w;
    // reduce within 8-lane head groups (wave32)
    p += __shfl_xor(p, 1, 32);
    p += __shfl_xor(p, 2, 32);
    p += __shfl_xor(p, 4, 32);
    if ((lane & 7) == 0) {
        const int h = lane >> 3;
        scores[(size_t)e * NHEAD + h] = p;
        atomicMax(&mbits[d * NHEAD + h], enc_f32(p));
    }
}

// ---------------- exp(score - max) and segment sum -----------------------------
__global__ __launch_bounds__(256)
void edge_softmax_kernel(float* __restrict__ scores, const int* __restrict__ dst,
                         const int* __restrict__ mbits, float* __restrict__ denom,
                         int ne)
{
    const int i = blockIdx.x * 256 + threadIdx.x;
    if (i >= ne * NHEAD) return;
    const int e = i >> 2, h = i & 3;
    const int d = dst[e];
    const float mv = dec_f32(mbits[d * NHEAD + h]);
    const float p  = expf(scores[i] - mv);
    scores[i] = p;                       // overwrite in place with numerator
    atomAddF32(&denom[d * NHEAD + h], p);
}

// ---------------- weighted scatter aggregation ---------------------------------
__global__ __launch_bounds__(256)
void edge_aggregate_kernel(const float* __restrict__ vf, const float* __restrict__ p,
                           const float* __restrict__ denom,
                           const int* __restrict__ src, const int* __restrict__ dst,
                           float* __restrict__ out, int ne)
{
    const int e = blockIdx.x * 8 + (threadIdx.x >> 5);
    if (e >= ne) return;
    const int lane = threadIdx.x & 31;
    const int s = src[e], d = dst[e];
    const int h = lane >> 3;
    const float alpha = p[(size_t)e * NHEAD + h] / denom[d * NHEAD + h];
    const float4 vv = *(const float4*)(vf + (size_t)s * DCH + lane * 4);
    float* o = out + (size_t)d * DCH + lane * 4;
    atomAddF32(o + 0, vv.x * alpha);
    atomAddF32(o + 1, vv.y * alpha);
    atomAddF32(o + 2, vv.z * alpha);
    atomAddF32(o + 3, vv.w * alpha);
}

// ---------------- launch -------------------------------------------------------
extern "C" void kernel_launch(void* const* d_in, const int* in_sizes, int n_in,
                              void* d_out, int out_size, void* d_ws, size_t ws_size,
                              hipStream_t stream)
{
    const float* x  = (const float*)d_in[0];
    const float* Wk = (const float*)d_in[1];
    const float* bk = (const float*)d_in[2];
    const float* Wq = (const float*)d_in[3];
    const float* bq = (const float*)d_in[4];
    const float* Wv = (const float*)d_in[5];
    const float* bv = (const float*)d_in[6];
    const int*  src = (const int*)d_in[7];
    const int*  dst = (const int*)d_in[8];
    const int n  = in_sizes[0] / DCH;
    const int ne = in_sizes[7];

    float* kf     = (float*)d_ws;
    float* qf     = kf + (size_t)n * DCH;
    float* vf     = qf + (size_t)n * DCH;
    float* scores = vf + (size_t)n * DCH;
    int*   mbits  = (int*)(scores + (size_t)ne * NHEAD);
    float* denom  = (float*)(mbits + (size_t)n * NHEAD);
    float* out    = (float*)d_out;

    init_kernel<<<(n * DCH + 255) / 256, 256, 0, stream>>>(mbits, denom, out, n);

    dim3 gg((n + DCH - 1) / DCH, 1, 3);
    proj_gemm<<<gg, 256, 0, stream>>>(x, Wk, bk, Wq, bq, Wv, bv, kf, qf, vf, n);

    edge_scores_kernel<<<(ne + 7) / 8, 256, 0, stream>>>(kf, qf, src, dst,
                                                         scores, mbits, ne);
    edge_softmax_kernel<<<(ne * NHEAD + 255) / 256, 256, 0, stream>>>(scores, dst,
                                                                      mbits, denom, ne);
    edge_aggregate_kernel<<<(ne + 7) / 8, 256, 0, stream>>>(vf, scores, denom,
                                                            src, dst, out, ne);
}